// SAModule_54365696033246
// MI455X (gfx1250) — compile-verified
//
#include <hip/hip_runtime.h>

typedef float v2f __attribute__((ext_vector_type(2)));
typedef float v8f __attribute__((ext_vector_type(8)));
typedef int v4i __attribute__((vector_size(16)));   // matches builtin param type

#define DEVFN static __device__ __forceinline__

constexpr int BATCH = 16;
constexpr int NPTS  = 4096;
constexpr int NCENT = 1024;                 // S
constexpr int KNB   = 64;                   // neighbors per center
constexpr int NGRP  = BATCH * NCENT;        // 16384
constexpr int NROWSTOT = NGRP * KNB;        // 1048576
constexpr float RAD2 = 0.2f * 0.2f;
constexpr float BN_EPS = 1e-5f;
constexpr float INV_NR = 1.0f / 1048576.0f;

// ---------------------------------------------------------------------------
// Async global->LDS copy path (CDNA5 GLOBAL_LOAD_ASYNC_TO_LDS_B128, ASYNCcnt).
// Builtin signature (from hipcc diagnostic): (v4i AS1*, v4i AS3*, Ii, Ii).
// Guarded by __has_builtin so the fallback keeps the verified synchronous path.
// ---------------------------------------------------------------------------
#if defined(__has_builtin)
#if __has_builtin(__builtin_amdgcn_global_load_async_to_lds_b128)
#define HAVE_ASYNC_LDS 1
#endif
#endif
#ifndef HAVE_ASYNC_LDS
#define HAVE_ASYNC_LDS 0
#endif

typedef __attribute__((address_space(1))) v4i* as1_v4i;
typedef __attribute__((address_space(3))) v4i* as3_v4i;

DEVFN void g2l_b128(float* lds_dst, const float* src) {
#if HAVE_ASYNC_LDS
  // Flat shared addresses carry the LDS byte offset in their low 32 bits
  // (ISA aperture rule); AS1 values equal flat.
  as3_v4i l = (as3_v4i)(unsigned long)(unsigned)(unsigned long)lds_dst;
  as1_v4i g = (as1_v4i)(unsigned long)src;
  __builtin_amdgcn_global_load_async_to_lds_b128(g, l, 0, 0);
#else
  *reinterpret_cast<float4*>(lds_dst) = *reinterpret_cast<const float4*>(src);
#endif
}

DEVFN void async_wait0() {
#if HAVE_ASYNC_LDS
#if __has_builtin(__builtin_amdgcn_s_wait_asynccnt)
  __builtin_amdgcn_s_wait_asynccnt(0);
#else
  asm volatile("s_wait_asynccnt 0x0" ::: "memory");
#endif
#endif
}

// ---------------------------------------------------------------------------
// WMMA helpers (V_WMMA_F32_16X16X4_F32, wave32)
// ---------------------------------------------------------------------------
DEVFN v8f wmma4(v2f a, v2f b, v8f c) {
  return __builtin_amdgcn_wmma_f32_16x16x4_f32(false, a, false, b, (short)0, c,
                                               false, false);
}

DEVFN v8f vzero8() {
  v8f z;
#pragma unroll
  for (int i = 0; i < 8; ++i) z[i] = 0.0f;
  return z;
}

// ---------------------------------------------------------------------------
// Kernel 1: farthest point sampling. One block per batch, 256 threads,
// 16 points per thread held entirely in registers.
// ---------------------------------------------------------------------------
__global__ __launch_bounds__(256) void fps_kernel(const float* __restrict__ pos,
                                                  int* __restrict__ cent) {
  __shared__ float rv[8];
  __shared__ int ri[8];
  __shared__ float scx, scy, scz;
  __shared__ int sfar;

  const int b = blockIdx.x;
  const float* p = pos + (size_t)b * NPTS * 3;
  const int base = threadIdx.x * 16;

  float X[16], Y[16], Z[16], D[16];
#pragma unroll
  for (int j = 0; j < 16; ++j) {
    int n = base + j;
    X[j] = p[n * 3 + 0];
    Y[j] = p[n * 3 + 1];
    Z[j] = p[n * 3 + 2];
    D[j] = 1e10f;
  }
  if (threadIdx.x == 0) { sfar = 0; scx = p[0]; scy = p[1]; scz = p[2]; }
  __syncthreads();

  const int lane = threadIdx.x & 31;
  const int wv = threadIdx.x >> 5;

  for (int it = 0; it < NCENT; ++it) {
    const float cx = scx, cy = scy, cz = scz;
    if (threadIdx.x == 0) cent[b * NCENT + it] = sfar;

    float bestv = -1.0f; int besti = 0;
#pragma unroll
    for (int j = 0; j < 16; ++j) {
      float dx = __fsub_rn(X[j], cx);
      float dy = __fsub_rn(Y[j], cy);
      float dz = __fsub_rn(Z[j], cz);
      float d = __fadd_rn(__fadd_rn(__fmul_rn(dx, dx), __fmul_rn(dy, dy)),
                          __fmul_rn(dz, dz));
      d = fminf(D[j], d);
      D[j] = d;
      if (d > bestv) { bestv = d; besti = base + j; }
    }
#pragma unroll
    for (int off = 16; off > 0; off >>= 1) {
      float ov = __shfl_down(bestv, off, 32);
      int oi = __shfl_down(besti, off, 32);
      if (ov > bestv || (ov == bestv && oi < besti)) { bestv = ov; besti = oi; }
    }
    if (lane == 0) { rv[wv] = bestv; ri[wv] = besti; }
    __syncthreads();
    if (threadIdx.x == 0) {
      float bv = rv[0]; int bi = ri[0];
      for (int w2 = 1; w2 < 8; ++w2)
        if (rv[w2] > bv || (rv[w2] == bv && ri[w2] < bi)) { bv = rv[w2]; bi = ri[w2]; }
      sfar = bi;
    }
    __syncthreads();
    if ((sfar >> 4) == (int)threadIdx.x) {
      int j = sfar & 15;
      scx = X[j]; scy = Y[j]; scz = Z[j];
    }
    __syncthreads();
  }
}

// ---------------------------------------------------------------------------
// Kernel 2: radius-ball grouping. One wave32 per center; ballot prefix
// compaction of in-ball indices (first KNB, padded with first index).
// ---------------------------------------------------------------------------
__global__ __launch_bounds__(128) void group_kernel(const float* __restrict__ pos,
                                                    const int* __restrict__ cent,
                                                    int* __restrict__ gidx,
                                                    float* __restrict__ cpos) {
  const int g = blockIdx.x * 4 + (threadIdx.x >> 5);
  const int lane = threadIdx.x & 31;
  const int b = g >> 10;  // NCENT = 1024
  const float* p = pos + (size_t)b * NPTS * 3;
  const int cid = cent[g];
  const float cx = p[cid * 3 + 0], cy = p[cid * 3 + 1], cz = p[cid * 3 + 2];
  if (lane == 0) {
    cpos[g * 3 + 0] = cx; cpos[g * 3 + 1] = cy; cpos[g * 3 + 2] = cz;
  }
  const float cc = __fadd_rn(__fadd_rn(__fmul_rn(cx, cx), __fmul_rn(cy, cy)),
                             __fmul_rn(cz, cz));
  int* out = gidx + (size_t)g * KNB;
  int total = 0;
  int first = 0;
  bool havefirst = false;

  for (int basep = 0; basep < NPTS; basep += 32) {
    if (total >= KNB) break;
    const int n = basep + lane;
    const float qx = p[n * 3 + 0], qy = p[n * 3 + 1], qz = p[n * 3 + 2];
    const float dot = __fadd_rn(__fadd_rn(__fmul_rn(cx, qx), __fmul_rn(cy, qy)),
                                __fmul_rn(cz, qz));
    const float pp = __fadd_rn(__fadd_rn(__fmul_rn(qx, qx), __fmul_rn(qy, qy)),
                               __fmul_rn(qz, qz));
    const float sq = __fadd_rn(__fsub_rn(cc, __fmul_rn(2.0f, dot)), pp);
    const bool in = (sq <= RAD2);
    const unsigned mask = (unsigned)__ballot(in);
    const int prefix = __popc(mask & ((1u << lane) - 1u));
    if (in) {
      int slot = total + prefix;
      if (slot < KNB) out[slot] = n;
    }
    if (!havefirst && mask != 0u) { first = basep + __builtin_ctz(mask); havefirst = true; }
    total += __popc(mask);
  }
  if (total < KNB) {
    for (int j = total + lane; j < KNB; j += 32) out[j] = first;
  }
}

// ---------------------------------------------------------------------------
// Shared staging helper: stage 16 gathered rows (feat CPAD=64, geo CPAD=8)
// into LDS for WMMA A-fragment reads. Feat path uses async global->LDS.
// ---------------------------------------------------------------------------
template <int CPAD>
DEVFN void stage_tile(int lane, int row0, const float* __restrict__ feat,
                      const float* __restrict__ pos, const int* __restrict__ gidx,
                      const float* __restrict__ cpos, float* xs) {
  if constexpr (CPAD == 64) {
    const int r = lane >> 1;
    const int half = lane & 1;
    const int row = row0 + r;
    const int b = row >> 16;  // NCENT*KNB = 65536 rows per batch
    const int idx = gidx[row];
    const float* src = feat + ((size_t)b * NPTS + idx) * 64 + half * 32;
    float* dst = xs + r * 64 + half * 32;
#pragma unroll
    for (int i = 0; i < 32; i += 4) g2l_b128(dst + i, src + i);
  } else {
    if (lane < 16) {
      const int row = row0 + lane;
      const int b = row >> 16;
      const int g = row >> 6;
      const int idx = gidx[row];
      const float* pp = pos + ((size_t)b * NPTS + idx) * 3;
      const float qx = pp[0], qy = pp[1], qz = pp[2];
      const float cx = cpos[g * 3 + 0], cy = cpos[g * 3 + 1], cz = cpos[g * 3 + 2];
      float* dst = xs + lane * 8;
      dst[0] = qx; dst[1] = qy; dst[2] = qz;
      dst[3] = qx - cx; dst[4] = qy - cy; dst[5] = qz - cz;
      dst[6] = 0.0f; dst[7] = 0.0f;
    }
  }
}

template <int CPAD>
DEVFN void conv1_frags(int lane, const float* xs, const float* W0, v8f d[4]) {
  const int l15 = lane & 15;
  const int half = lane >> 4;
#pragma unroll
  for (int nt = 0; nt < 4; ++nt) d[nt] = vzero8();
#pragma unroll
  for (int kk = 0; kk < CPAD; kk += 4) {
    v2f a = *reinterpret_cast<const v2f*>(xs + l15 * CPAD + kk + 2 * half);
#pragma unroll
    for (int nt = 0; nt < 4; ++nt) {
      v2f bw = *reinterpret_cast<const v2f*>(W0 + (nt * 16 + l15) * CPAD + kk + 2 * half);
      d[nt] = wmma4(a, bw, d[nt]);
    }
  }
}

DEVFN void write_y(int lane, const v8f d[4], const float bb0[4], float* ydst) {
  const int l15 = lane & 15;
  const int half = lane >> 4;
#pragma unroll
  for (int nt = 0; nt < 4; ++nt)
#pragma unroll
    for (int r = 0; r < 8; ++r) {
      float v = fmaxf(d[nt][r] + bb0[nt], 0.0f);
      ydst[(r + 8 * half) * 64 + nt * 16 + l15] = v;
    }
}

DEVFN void conv2_frags(int lane, const float* ys, const float* W1, v8f z[8]) {
  const int l15 = lane & 15;
  const int half = lane >> 4;
#pragma unroll
  for (int nt = 0; nt < 8; ++nt) z[nt] = vzero8();
#pragma unroll
  for (int kk = 0; kk < 64; kk += 4) {
    v2f a = *reinterpret_cast<const v2f*>(ys + l15 * 64 + kk + 2 * half);
#pragma unroll
    for (int nt = 0; nt < 8; ++nt) {
      v2f bw = *reinterpret_cast<const v2f*>(W1 + (nt * 16 + l15) * 64 + kk + 2 * half);
      z[nt] = wmma4(a, bw, z[nt]);
    }
  }
}

// ---------------------------------------------------------------------------
// Kernel 3: input moments (sum x, sum x x^T) of gathered rows. MODE 0 = feat
// (C=64), MODE 1 = geo (C=8 padded, cols 6/7 zero).
// ---------------------------------------------------------------------------
template <int C, int MODE>
__global__ __launch_bounds__(256) void moments0_kernel(
    const float* __restrict__ feat, const float* __restrict__ pos,
    const int* __restrict__ gidx, const float* __restrict__ cpos,
    float* __restrict__ Sx, float* __restrict__ Sxx) {
  __shared__ alignas(16) float Xs[64 * C];
  constexpr int NENT = C * C + C;
  constexpr int NACC = (NENT + 255) / 256;
  float acc[NACC];
#pragma unroll
  for (int i = 0; i < NACC; ++i) acc[i] = 0.0f;

  const int nchunks = NROWSTOT / 64;
  for (int chunk = blockIdx.x; chunk < nchunks; chunk += gridDim.x) {
    const int row0 = chunk * 64;
    const int nxt = chunk + gridDim.x;
    if (nxt < nchunks) __builtin_prefetch(&gidx[nxt * 64], 0, 1);
    if constexpr (MODE == 0) {
      const int r = threadIdx.x >> 2;
      const int part = threadIdx.x & 3;
      const int row = row0 + r;
      const int b = row >> 16;
      const int idx = gidx[row];
      const float* src = feat + ((size_t)b * NPTS + idx) * 64 + part * 16;
      float* dst = Xs + r * 64 + part * 16;
#pragma unroll
      for (int i = 0; i < 16; i += 4) g2l_b128(dst + i, src + i);
      async_wait0();
    } else {
      if (threadIdx.x < 64) {
        const int row = row0 + (int)threadIdx.x;
        const int b = row >> 16;
        const int g = row >> 6;
        const int idx = gidx[row];
        const float* pp = pos + ((size_t)b * NPTS + idx) * 3;
        const float qx = pp[0], qy = pp[1], qz = pp[2];
        const float cx = cpos[g * 3 + 0], cy = cpos[g * 3 + 1], cz = cpos[g * 3 + 2];
        float* dst = Xs + threadIdx.x * 8;
        dst[0] = qx; dst[1] = qy; dst[2] = qz;
        dst[3] = qx - cx; dst[4] = qy - cy; dst[5] = qz - cz;
        dst[6] = 0.0f; dst[7] = 0.0f;
      }
    }
    __syncthreads();
    int k = 0;
    for (int e = threadIdx.x; e < NENT; e += 256, ++k) {
      float a = acc[k];
      if (e < C * C) {
        const int i = e / C, j = e % C;
        for (int r = 0; r < 64; ++r) a += Xs[r * C + i] * Xs[r * C + j];
      } else {
        const int i = e - C * C;
        for (int r = 0; r < 64; ++r) a += Xs[r * C + i];
      }
      acc[k] = a;
    }
    __syncthreads();
  }
  int k = 0;
  for (int e = threadIdx.x; e < NENT; e += 256, ++k) {
    if (e < C * C) atomicAdd(&Sxx[e], acc[k]);
    else atomicAdd(&Sx[e - C * C], acc[k]);
  }
}

// ---------------------------------------------------------------------------
// Kernel 4: fold layer-0 BN into weights using derived stats.
// ---------------------------------------------------------------------------
__global__ __launch_bounds__(64) void fold0_kernel(
    const float* __restrict__ w, const float* __restrict__ bias,
    const float* __restrict__ gm, const float* __restrict__ bt,
    const float* __restrict__ Sx, const float* __restrict__ Sxx,
    float* __restrict__ wf, float* __restrict__ bf, int Cin, int Cpad) {
  const int o = threadIdx.x;
  if (o >= 64) return;
  float m = bias[o];
  for (int i = 0; i < Cin; ++i) m += w[o * Cin + i] * Sx[i] * INV_NR;
  float q = 0.0f;
  for (int i = 0; i < Cin; ++i) {
    float t = 0.0f;
    for (int j = 0; j < Cin; ++j) t += Sxx[i * Cpad + j] * INV_NR * w[o * Cin + j];
    q += w[o * Cin + i] * t;
  }
  const float b = bias[o];
  q += 2.0f * b * (m - b) + b * b;
  const float var = q - m * m;
  const float a = gm[o] * rsqrtf(var + BN_EPS);
  const float c = bt[o] - m * a;
  for (int i = 0; i < Cpad; ++i) wf[o * Cpad + i] = (i < Cin) ? w[o * Cin + i] * a : 0.0f;
  bf[o] = b * a + c;
}

// ---------------------------------------------------------------------------
// Kernel 5: y = relu(conv1(x)) via WMMA + accumulate moments of y (C=64).
// ---------------------------------------------------------------------------
template <int CPAD>
__global__ __launch_bounds__(128) void m1_kernel(
    const float* __restrict__ feat, const float* __restrict__ pos,
    const int* __restrict__ gidx, const float* __restrict__ cpos,
    const float* __restrict__ w0f, const float* __restrict__ b0f,
    float* __restrict__ Sy, float* __restrict__ Syy) {
  __shared__ alignas(16) float W0[64 * CPAD];
  __shared__ alignas(16) float Xs[4][16 * CPAD];
  __shared__ alignas(16) float Ys[64 * 64];

  for (int i = threadIdx.x; i < 64 * CPAD; i += 128) W0[i] = w0f[i];
  const int lane = threadIdx.x & 31;
  const int wv = threadIdx.x >> 5;
  float bb0[4];
#pragma unroll
  for (int nt = 0; nt < 4; ++nt) bb0[nt] = b0f[nt * 16 + (lane & 15)];

  constexpr int NENT = 64 * 64 + 64;
  constexpr int NACC = (NENT + 127) / 128;
  float acc[NACC];
#pragma unroll
  for (int i = 0; i < NACC; ++i) acc[i] = 0.0f;
  __syncthreads();

  const int nchunks = NROWSTOT / 64;
  for (int chunk = blockIdx.x; chunk < nchunks; chunk += gridDim.x) {
    const int row0 = chunk * 64 + wv * 16;
    const int nxt = chunk + gridDim.x;
    if (nxt < nchunks) __builtin_prefetch(&gidx[nxt * 64 + wv * 16], 0, 1);
    stage_tile<CPAD>(lane, row0, feat, pos, gidx, cpos, Xs[wv]);
    async_wait0();
    v8f d[4];
    conv1_frags<CPAD>(lane, Xs[wv], W0, d);
    write_y(lane, d, bb0, &Ys[wv * 16 * 64]);
    __syncthreads();
    int k = 0;
    for (int e = threadIdx.x; e < NENT; e += 128, ++k) {
      float a = acc[k];
      if (e < 4096) {
        const int i = e >> 6, j = e & 63;
        for (int r = 0; r < 64; ++r) a += Ys[r * 64 + i] * Ys[r * 64 + j];
      } else {
        const int i = e - 4096;
        for (int r = 0; r < 64; ++r) a += Ys[r * 64 + i];
      }
      acc[k] = a;
    }
    __syncthreads();
  }
  int k = 0;
  for (int e = threadIdx.x; e < NENT; e += 128, ++k) {
    if (e < 4096) atomicAdd(&Syy[e], acc[k]);
    else atomicAdd(&Sy[e - 4096], acc[k]);
  }
}

// ---------------------------------------------------------------------------
// Kernel 6: fold layer-1 BN into weights (Cout = 128, Cin = 64).
// ---------------------------------------------------------------------------
__global__ __launch_bounds__(128) void fold1_kernel(
    const float* __restrict__ w, const float* __restrict__ bias,
    const float* __restrict__ gm, const float* __restrict__ bt,
    const float* __restrict__ Sy, const float* __restrict__ Syy,
    float* __restrict__ wf, float* __restrict__ bf) {
  const int o = threadIdx.x;
  if (o >= 128) return;
  float m = bias[o];
  for (int i = 0; i < 64; ++i) m += w[o * 64 + i] * Sy[i] * INV_NR;
  float q = 0.0f;
  for (int i = 0; i < 64; ++i) {
    float t = 0.0f;
    for (int j = 0; j < 64; ++j) t += Syy[i * 64 + j] * INV_NR * w[o * 64 + j];
    q += w[o * 64 + i] * t;
  }
  const float b = bias[o];
  q += 2.0f * b * (m - b) + b * b;
  const float var = q - m * m;
  const float a = gm[o] * rsqrtf(var + BN_EPS);
  const float c = bt[o] - m * a;
  for (int i = 0; i < 64; ++i) wf[o * 64 + i] = w[o * 64 + i] * a;
  bf[o] = b * a + c;
}

// ---------------------------------------------------------------------------
// Kernel 7: fused final pass. One wave per (b,s) group: gather -> WMMA conv1
// -> relu -> LDS transpose -> WMMA conv2 -> relu -> max over K -> out.
// X staging buffer is reused as Y staging (per-wave DS ops are in-order).
// ---------------------------------------------------------------------------
template <int CPAD>
__global__ __launch_bounds__(128) void final_kernel(
    const float* __restrict__ feat, const float* __restrict__ pos,
    const int* __restrict__ gidx, const float* __restrict__ cpos,
    const float* __restrict__ w0f, const float* __restrict__ b0f,
    const float* __restrict__ w1f, const float* __restrict__ b1f,
    float* __restrict__ out) {
  __shared__ alignas(16) float W0[64 * CPAD];
  __shared__ alignas(16) float W1[128 * 64];
  __shared__ alignas(16) float XY[4][16 * 64];                   // X then Y staging
  __shared__ alignas(16) float XsG[4][(CPAD == 8) ? 16 * 8 : 4]; // geo-only X staging

  for (int i = threadIdx.x; i < 64 * CPAD; i += 128) W0[i] = w0f[i];
  for (int i = threadIdx.x; i < 128 * 64; i += 128) W1[i] = w1f[i];
  const int lane = threadIdx.x & 31;
  const int wv = threadIdx.x >> 5;
  const int g = blockIdx.x * 4 + wv;
  float bb0[4], bb1[8];
#pragma unroll
  for (int nt = 0; nt < 4; ++nt) bb0[nt] = b0f[nt * 16 + (lane & 15)];
#pragma unroll
  for (int nt = 0; nt < 8; ++nt) bb1[nt] = b1f[nt * 16 + (lane & 15)];
  float mx[8];
#pragma unroll
  for (int nt = 0; nt < 8; ++nt) mx[nt] = -3.4e38f;
  __syncthreads();

  for (int tile = 0; tile < 4; ++tile) {
    const int row0 = g * 64 + tile * 16;
    float* xstage = (CPAD == 64) ? XY[wv] : XsG[wv];
    stage_tile<CPAD>(lane, row0, feat, pos, gidx, cpos, xstage);
    async_wait0();
    v8f d[4];
    conv1_frags<CPAD>(lane, xstage, W0, d);          // reads X
    write_y(lane, d, bb0, XY[wv]);                   // overwrites with Y
    v8f z[8];
    conv2_frags(lane, XY[wv], W1, z);
#pragma unroll
    for (int nt = 0; nt < 8; ++nt) {
      float m = mx[nt];
#pragma unroll
      for (int r = 0; r < 8; ++r) m = fmaxf(m, fmaxf(z[nt][r] + bb1[nt], 0.0f));
      mx[nt] = m;
    }
  }
  // merge the two M-halves (lane and lane^16 hold complementary row halves)
#pragma unroll
  for (int nt = 0; nt < 8; ++nt) {
    float o = __shfl_xor(mx[nt], 16, 32);
    mx[nt] = fmaxf(mx[nt], o);
  }
  if (lane < 16) {
#pragma unroll
    for (int nt = 0; nt < 8; ++nt)
      out[(size_t)g * 128 + nt * 16 + lane] = mx[nt];
  }
}

// ---------------------------------------------------------------------------
// Host launcher
// ---------------------------------------------------------------------------
extern "C" void kernel_launch(void* const* d_in, const int* in_sizes, int n_in,
                              void* d_out, int out_size, void* d_ws, size_t ws_size,
                              hipStream_t stream) {
  (void)in_sizes; (void)n_in; (void)out_size; (void)ws_size;
  const float* pos  = (const float*)d_in[0];
  const float* feat = (const float*)d_in[1];
  const float* w0   = (const float*)d_in[2];
  const float* b0   = (const float*)d_in[3];
  const float* gm0  = (const float*)d_in[4];
  const float* bt0  = (const float*)d_in[5];
  const float* w1   = (const float*)d_in[6];
  const float* b1   = (const float*)d_in[7];
  const float* gm1  = (const float*)d_in[8];
  const float* bt1  = (const float*)d_in[9];
  const float* wg0  = (const float*)d_in[10];
  const float* bg0  = (const float*)d_in[11];
  const float* gmg0 = (const float*)d_in[12];
  const float* btg0 = (const float*)d_in[13];
  const float* wg1  = (const float*)d_in[14];
  const float* bg1  = (const float*)d_in[15];
  const float* gmg1 = (const float*)d_in[16];
  const float* btg1 = (const float*)d_in[17];
  float* out = (float*)d_out;

  char* w = (char*)d_ws;
  int* cent = (int*)w;            w += (size_t)NGRP * sizeof(int);
  int* gidx = (int*)w;            w += (size_t)NROWSTOT * sizeof(int);
  float* cpos = (float*)w;        w += (size_t)NGRP * 3 * sizeof(float);
  float* stats = (float*)w;
  float* SxF  = stats;
  float* SxxF = SxF + 64;
  float* SyF  = SxxF + 4096;
  float* SyyF = SyF + 64;
  float* SxG  = SyyF + 4096;
  float* SxxG = SxG + 8;
  float* SyG  = SxxG + 64;
  float* SyyG = SyG + 64;
  const size_t nstat = 64 + 4096 + 64 + 4096 + 8 + 64 + 64 + 4096;
  w += nstat * sizeof(float);
  float* w0fF = (float*)w; w += 64 * 64 * sizeof(float);
  float* b0fF = (float*)w; w += 64 * sizeof(float);
  float* w1fF = (float*)w; w += 128 * 64 * sizeof(float);
  float* b1fF = (float*)w; w += 128 * sizeof(float);
  float* w0fG = (float*)w; w += 64 * 8 * sizeof(float);
  float* b0fG = (float*)w; w += 64 * sizeof(float);
  float* w1fG = (float*)w; w += 128 * 64 * sizeof(float);
  float* b1fG = (float*)w; w += 128 * sizeof(float);

  (void)hipMemsetAsync(stats, 0, nstat * sizeof(float), stream);

  fps_kernel<<<BATCH, 256, 0, stream>>>(pos, cent);
  group_kernel<<<NGRP / 4, 128, 0, stream>>>(pos, cent, gidx, cpos);

  moments0_kernel<64, 0><<<1024, 256, 0, stream>>>(feat, pos, gidx, cpos, SxF, SxxF);
  moments0_kernel<8, 1><<<1024, 256, 0, stream>>>(feat, pos, gidx, cpos, SxG, SxxG);
  fold0_kernel<<<1, 64, 0, stream>>>(w0, b0, gm0, bt0, SxF, SxxF, w0fF, b0fF, 64, 64);
  fold0_kernel<<<1, 64, 0, stream>>>(wg0, bg0, gmg0, btg0, SxG, SxxG, w0fG, b0fG, 6, 8);

  m1_kernel<64><<<2048, 128, 0, stream>>>(feat, pos, gidx, cpos, w0fF, b0fF, SyF, SyyF);
  m1_kernel<8><<<2048, 128, 0, stream>>>(feat, pos, gidx, cpos, w0fG, b0fG, SyG, SyyG);
  fold1_kernel<<<1, 128, 0, stream>>>(w1, b1, gm1, bt1, SyF, SyyF, w1fF, b1fF);
  fold1_kernel<<<1, 128, 0, stream>>>(wg1, bg1, gmg1, btg1, SyG, SyyG, w1fG, b1fG);

  final_kernel<64><<<NGRP / 4, 128, 0, stream>>>(feat, pos, gidx, cpos,
                                                 w0fF, b0fF, w1fF, b1fF, out);
  final_kernel<8><<<NGRP / 4, 128, 0, stream>>>(feat, pos, gidx, cpos,
                                                w0fG, b0fG, w1fG, b1fG,
                                                out + (size_t)NGRP * 128);
}